// _KANLinear_32375463477685
// MI455X (gfx1250) — compile-verified
//
#include <hip/hip_runtime.h>

// KAN linear layer fused as one f32 GEMM:
//   out[b,o] = sum_i ( silu(x[b,i]) * base_w[o,i]
//                    + sum_n bases(tanh(x[b,i]))[n] * spline_w[o,i,n] )
// M=4096, N=1024, K = 1024 channels * 9 values/channel = 9216.
// V_WMMA_F32_16X16X4_F32 (exact f32 precision). BM=64 x BN=256 tiles halve
// the spline-basis VALU recompute vs BN=128; weights stay L2-resident.

typedef __attribute__((ext_vector_type(2))) float v2f;
typedef __attribute__((ext_vector_type(8))) float v8f;

#define IN_F   1024
#define OUT_F  1024
#define BM     64
#define BN     256
#define IC     8          // input channels staged per K-chunk
#define KC     (IC * 9)   // 72 K-values per chunk
#define LDA    74         // padded LDS row stride (floats); even -> 8B-aligned float2
#define LDB    74
#define NTHREADS 256
#define AQ     ((BM * IC) / NTHREADS)   // 2 x-elements per thread
#define BQ     ((BN * IC) / NTHREADS)   // 8 weight pairs per thread

// Order-3 Cox-de Boor recursion on uniform grid: g[t] = -2.2 + 0.4*t, t=0..11.
// Denominators are constant (0.4, 0.8, 1.2) + 1e-8, matching the reference.
__device__ __forceinline__ void kan_bases(float xt, float bs[8]) {
    float b0[11];
#pragma unroll
    for (int t = 0; t < 11; ++t) {
        float gl = -2.2f + 0.4f * (float)t;
        b0[t] = (xt >= gl && xt < gl + 0.4f) ? 1.0f : 0.0f;
    }
    const float r1 = 1.0f / (0.4f + 1e-8f);
    const float r2 = 1.0f / (0.8f + 1e-8f);
    const float r3 = 1.0f / (1.2f + 1e-8f);
    float b1[10];
#pragma unroll
    for (int t = 0; t < 10; ++t) {
        float gt = -2.2f + 0.4f * (float)t;
        b1[t] = (xt - gt) * r1 * b0[t] + ((gt + 0.8f) - xt) * r1 * b0[t + 1];
    }
    float b2[9];
#pragma unroll
    for (int t = 0; t < 9; ++t) {
        float gt = -2.2f + 0.4f * (float)t;
        b2[t] = (xt - gt) * r2 * b1[t] + ((gt + 1.2f) - xt) * r2 * b1[t + 1];
    }
#pragma unroll
    for (int t = 0; t < 8; ++t) {
        float gt = -2.2f + 0.4f * (float)t;
        bs[t] = (xt - gt) * r3 * b2[t] + ((gt + 1.6f) - xt) * r3 * b2[t + 1];
    }
}

__global__ __launch_bounds__(NTHREADS)
void kan_fused_wmma_kernel(const float* __restrict__ x,
                           const float* __restrict__ spline_w,  // (O, I, 8)
                           const float* __restrict__ base_w,    // (O, I)
                           float* __restrict__ out) {           // (B, O)
    __shared__ float lA[BM * LDA];   // activation tile: [silu | 8 bases] per channel
    __shared__ float lB[BN * LDB];   // weight tile:     [base_w | 8 spline_w] per channel

    const int tid  = threadIdx.x;
    const int lane = tid & 31;
    const int wave = tid >> 5;
    const int l15  = lane & 15;
    const int khi  = (lane >> 4) * 2;   // half-wave K offset for 16x4 f32 A/B frags

    const int m_base = blockIdx.y * BM;
    const int n_base = blockIdx.x * BN;

    const int wm = wave & 1;    // 2 slabs of 32 rows   (BM = 64)
    const int wn = wave >> 1;   // 4 slabs of 64 cols   (BN = 256)

    v8f acc[2][4] = {};         // 2 M-tiles x 4 N-tiles of 16x16 f32

    // ---- register-pipelined x values (chunk 0 prologue) ----
    float xreg[AQ];
#pragma unroll
    for (int q = 0; q < AQ; ++q) {
        int e = tid + q * NTHREADS;
        xreg[q] = x[(size_t)(m_base + (e >> 3)) * IN_F + (e & 7)];
    }

    for (int i0 = 0; i0 < IN_F; i0 += IC) {
        // ---- stage A tile: silu + spline bases from pipelined registers ----
#pragma unroll
        for (int q = 0; q < AQ; ++q) {
            int e = tid + q * NTHREADS;
            int r = e >> 3;
            int c = e & 7;
            float xv = xreg[q];
            float xt = tanhf(xv);
            float bs[8];
            kan_bases(xt, bs);
            float* dst = &lA[r * LDA + c * 9];
            dst[0] = xv / (1.0f + __expf(-xv));   // silu
#pragma unroll
            for (int j = 0; j < 8; ++j) dst[1 + j] = bs[j];
        }
        // ---- stage B tile: base_w scalar + spline_w float4x2 per (o, channel) ----
#pragma unroll
        for (int q = 0; q < BQ; ++q) {
            int p = tid + q * NTHREADS;
            int o = p >> 3;
            int c = p & 7;
            int go = n_base + o;
            float bwv = base_w[(size_t)go * IN_F + i0 + c];
            const float4* sp =
                (const float4*)&spline_w[((size_t)go * IN_F + (i0 + c)) * 8];
            float4 s0 = sp[0];
            float4 s1 = sp[1];
            float* dst = &lB[o * LDB + c * 9];
            dst[0] = bwv;
            dst[1] = s0.x; dst[2] = s0.y; dst[3] = s0.z; dst[4] = s0.w;
            dst[5] = s1.x; dst[6] = s1.y; dst[7] = s1.z; dst[8] = s1.w;
        }
        __syncthreads();

        // ---- issue next-chunk x loads + weight prefetches under the WMMAs ----
        const int i0n = i0 + IC;
        if (i0n < IN_F) {
#pragma unroll
            for (int q = 0; q < AQ; ++q) {
                int e = tid + q * NTHREADS;
                xreg[q] = x[(size_t)(m_base + (e >> 3)) * IN_F + i0n + (e & 7)];
            }
            // prefetch next weight lines (global_prefetch_b8); L2 -> near caches
            {
                int p0 = tid;                       // covers an eighth of the spline lines (o 0..31)
                int p4 = tid + 4 * NTHREADS;        // covers another eighth (o 128..159)
                __builtin_prefetch(
                    &spline_w[((size_t)(n_base + (p0 >> 3)) * IN_F + i0n + (p0 & 7)) * 8], 0, 1);
                __builtin_prefetch(
                    &spline_w[((size_t)(n_base + (p4 >> 3)) * IN_F + i0n + (p4 & 7)) * 8], 0, 1);
                __builtin_prefetch(
                    &base_w[(size_t)(n_base + (p0 >> 3)) * IN_F + i0n + (p0 & 7)], 0, 1);
            }
        }

        // ---- MACs: 18 K-steps of V_WMMA_F32_16X16X4_F32, 8 tiles per wave ----
#pragma unroll
        for (int ks = 0; ks < KC / 4; ++ks) {
            int kk = ks * 4 + khi;
            v2f af[2], bf[4];
#pragma unroll
            for (int mt = 0; mt < 2; ++mt) {
                int row = wm * 32 + mt * 16 + l15;
                af[mt] = *(const v2f*)&lA[row * LDA + kk];
            }
#pragma unroll
            for (int nt = 0; nt < 4; ++nt) {
                int col = wn * 64 + nt * 16 + l15;
                bf[nt] = *(const v2f*)&lB[col * LDB + kk];
            }
#pragma unroll
            for (int mt = 0; mt < 2; ++mt)
#pragma unroll
                for (int nt = 0; nt < 4; ++nt)
                    acc[mt][nt] = __builtin_amdgcn_wmma_f32_16x16x4_f32(
                        false, af[mt], false, bf[nt],
                        (short)0, acc[mt][nt], false, false);
        }
        __syncthreads();
    }

    // ---- store: C/D layout = 8 VGPRs, lane half selects M+8, lane&15 = N ----
#pragma unroll
    for (int mt = 0; mt < 2; ++mt) {
        int m0 = m_base + wm * 32 + mt * 16 + (lane >> 4) * 8;
#pragma unroll
        for (int nt = 0; nt < 4; ++nt) {
            int n0 = n_base + wn * 64 + nt * 16 + l15;
#pragma unroll
            for (int j = 0; j < 8; ++j) {
                out[(size_t)(m0 + j) * OUT_F + n0] = acc[mt][nt][j];
            }
        }
    }
}

extern "C" void kernel_launch(void* const* d_in, const int* in_sizes, int n_in,
                              void* d_out, int out_size, void* d_ws, size_t ws_size,
                              hipStream_t stream) {
    const float* x   = (const float*)d_in[0];   // (B, 1024)
    const float* sw  = (const float*)d_in[1];   // (1024, 1024, 8)
    const float* bw  = (const float*)d_in[2];   // (1024, 1024)
    float* out = (float*)d_out;                 // (B, 1024)

    int batch = in_sizes[0] / IN_F;             // 4096
    dim3 grid(OUT_F / BN, batch / BM);          // (4, 64)
    kan_fused_wmma_kernel<<<grid, NTHREADS, 0, stream>>>(x, sw, bw, out);
}